// Gumbel24Linear_35330400977108
// MI455X (gfx1250) — compile-verified
//
#include <hip/hip_runtime.h>

typedef __attribute__((ext_vector_type(32))) _Float16 v32h;
typedef __attribute__((ext_vector_type(16))) _Float16 v16h;
typedef __attribute__((ext_vector_type(8)))  _Float16 v8h;
typedef __attribute__((ext_vector_type(4)))  _Float16 v4h;
typedef __attribute__((ext_vector_type(8)))  float    v8f;

#define IN_F   4096
#define OUT_F  4096
#define M_TOT  4096                    // 2 * 2048
#define N_BLOCKS (IN_F * OUT_F / 4)    // 4,194,304 4-wide blocks
#define KC     (IN_F / 2)              // compressed K per W row = 2048
#define NIDX32 (N_BLOCKS / 8)          // packed u32 index words

// ---------------------------------------------------------------------------
// Kernel 1: Gumbel argmax -> 2:4 compress weight (f16 pairs) + packed indices
//   one thread per 8 consecutive blocks => writes one u32 of 8 index nibbles
// ---------------------------------------------------------------------------
__global__ void g24_pack_w(const float* __restrict__ weight,
                           const float* __restrict__ cw,
                           const float* __restrict__ gum,
                           _Float16* __restrict__ wcomp,
                           unsigned* __restrict__ widx) {
    int t = blockIdx.x * blockDim.x + threadIdx.x;      // < NIDX32
    size_t b0 = (size_t)t * 8;                          // first 4-block

    float g[48], c[48];
    const float4* g4 = reinterpret_cast<const float4*>(gum + b0 * 6);
    const float4* c4 = reinterpret_cast<const float4*>(cw  + b0 * 6);
    #pragma unroll
    for (int p = 0; p < 12; ++p) {
        float4 gv = g4[p], cv = c4[p];
        g[4*p+0]=gv.x; g[4*p+1]=gv.y; g[4*p+2]=gv.z; g[4*p+3]=gv.w;
        c[4*p+0]=cv.x; c[4*p+1]=cv.y; c[4*p+2]=cv.z; c[4*p+3]=cv.w;
    }

    const int i0tab[6] = {2, 1, 1, 0, 0, 0};
    const int i1tab[6] = {3, 3, 2, 3, 2, 1};

    _Float16 comp[16];
    unsigned idx32 = 0u;
    const float4* w4p = reinterpret_cast<const float4*>(weight) + b0;

    #pragma unroll
    for (int bb = 0; bb < 8; ++bb) {
        int base = bb * 6;
        int idx = 0; float best = c[base] + g[base];
        #pragma unroll
        for (int j = 1; j < 6; ++j) {
            float l = c[base + j] + g[base + j];
            if (l > best) { best = l; idx = j; }
        }
        int i0 = i0tab[idx], i1 = i1tab[idx];

        float4 w4 = w4p[bb];
        float wa[4] = {w4.x, w4.y, w4.z, w4.w};
        comp[2*bb + 0] = (_Float16)wa[i0];
        comp[2*bb + 1] = (_Float16)wa[i1];
        idx32 |= (unsigned)((i1 << 2) | i0) << (4 * bb);
    }

    uint4* dst = reinterpret_cast<uint4*>(wcomp + b0 * 2);
    dst[0] = *reinterpret_cast<const uint4*>(&comp[0]);
    dst[1] = *reinterpret_cast<const uint4*>(&comp[8]);
    widx[t] = idx32;
}

// ---------------------------------------------------------------------------
// Kernel 2: x fp32 -> f16
// ---------------------------------------------------------------------------
__global__ void g24_quant_x(const float* __restrict__ x,
                            _Float16* __restrict__ xq) {
    int i = blockIdx.x * blockDim.x + threadIdx.x;
    float4 v = reinterpret_cast<const float4*>(x)[i];
    v4h o;
    o[0] = (_Float16)v.x; o[1] = (_Float16)v.y;
    o[2] = (_Float16)v.z; o[3] = (_Float16)v.w;
    *reinterpret_cast<v4h*>(xq + (size_t)i * 4) = o;
}

// ---------------------------------------------------------------------------
// Kernel 3: 2:4 sparse GEMM via V_SWMMAC_F32_16X16X64_F16
//   double-buffered LDS panels filled with GLOBAL_LOAD_ASYNC_TO_LDS (ASYNCcnt)
// ---------------------------------------------------------------------------
#define TN  128         // W feature rows per block
#define TMX 128         // x rows per block
#define TK  64          // expanded K per panel (32 compressed)

#define ASYNC_B128_OFF0(l, g)  asm volatile("global_load_async_to_lds_b128 %0, %1, off"           :: "v"(l), "v"(g) : "memory")
#define ASYNC_B128_OFF16(l, g) asm volatile("global_load_async_to_lds_b128 %0, %1, off offset:16" :: "v"(l), "v"(g) : "memory")
#define ASYNC_B128_OFF32(l, g) asm volatile("global_load_async_to_lds_b128 %0, %1, off offset:32" :: "v"(l), "v"(g) : "memory")
#define ASYNC_B128_OFF48(l, g) asm volatile("global_load_async_to_lds_b128 %0, %1, off offset:48" :: "v"(l), "v"(g) : "memory")
#define ASYNC_B32_OFF0(l, g)   asm volatile("global_load_async_to_lds_b32 %0, %1, off"            :: "v"(l), "v"(g) : "memory")
#define WAIT_ASYNC_7()         asm volatile("s_wait_asynccnt 7" ::: "memory")
#define WAIT_ASYNC_0()         asm volatile("s_wait_asynccnt 0" ::: "memory")

// issue one panel's staging as 7 async lane-copies (2xW, 1xidx, 4xX)
__device__ __forceinline__ void stage_async(
    const _Float16* __restrict__ wcomp, const unsigned* __restrict__ widx,
    const _Float16* __restrict__ xq,
    _Float16 (*Ws)[40], unsigned (*Is)[2], _Float16 (*Xs)[72],
    int tileN, int tileM, int k0, int ldr, int sw) {
    const int kc0 = k0 >> 1;
    unsigned lw = (unsigned)(uintptr_t)&Ws[ldr][sw * 16];
    unsigned li = (unsigned)(uintptr_t)&Is[ldr][sw];
    unsigned lx = (unsigned)(uintptr_t)&Xs[ldr][sw * 32];
    unsigned long long gw = (unsigned long long)(uintptr_t)
        (wcomp + (size_t)(tileN + ldr) * KC + kc0 + sw * 16);
    unsigned long long gi = (unsigned long long)(uintptr_t)
        (widx + (size_t)(tileN + ldr) * (IN_F / 32) + (k0 >> 5) + sw);
    unsigned long long gx = (unsigned long long)(uintptr_t)
        (xq + (size_t)(tileM + ldr) * IN_F + k0 + sw * 32);
    ASYNC_B128_OFF0 (lw, gw);
    ASYNC_B128_OFF16(lw, gw);
    ASYNC_B32_OFF0  (li, gi);
    ASYNC_B128_OFF0 (lx, gx);
    ASYNC_B128_OFF16(lx, gx);
    ASYNC_B128_OFF32(lx, gx);
    ASYNC_B128_OFF48(lx, gx);
}

// 8 SWMMACs on one resident panel
__device__ __forceinline__ void compute_panel(
    _Float16 (*Ws)[40], unsigned (*Is)[2], _Float16 (*Xs)[72],
    int waveA, int waveB, int m16, int hi, v8f (&acc)[4][2]) {
    v16h afrag[4];
    unsigned idxreg[4];
    #pragma unroll
    for (int i = 0; i < 4; ++i) {
        int row = waveA * 64 + i * 16 + m16;
        int ka  = hi * 8;
        v8h lo = *reinterpret_cast<const v8h*>(&Ws[row][ka]);
        v8h hh = *reinterpret_cast<const v8h*>(&Ws[row][ka + 16]);
        afrag[i] = __builtin_shufflevector(lo, hh,
            0,1,2,3,4,5,6,7,8,9,10,11,12,13,14,15);
        idxreg[i] = Is[row][hi];
    }
    v32h bfrag[2];
    #pragma unroll
    for (int j = 0; j < 2; ++j) {
        int col = waveB * 32 + j * 16 + m16;
        int kb  = hi * 16;
        v8h b0 = *reinterpret_cast<const v8h*>(&Xs[col][kb + 0]);
        v8h b1 = *reinterpret_cast<const v8h*>(&Xs[col][kb + 8]);
        v8h b2 = *reinterpret_cast<const v8h*>(&Xs[col][kb + 32]);
        v8h b3 = *reinterpret_cast<const v8h*>(&Xs[col][kb + 40]);
        v16h lo = __builtin_shufflevector(b0, b1,
            0,1,2,3,4,5,6,7,8,9,10,11,12,13,14,15);
        v16h hh = __builtin_shufflevector(b2, b3,
            0,1,2,3,4,5,6,7,8,9,10,11,12,13,14,15);
        bfrag[j] = __builtin_shufflevector(lo, hh,
            0,1,2,3,4,5,6,7,8,9,10,11,12,13,14,15,
            16,17,18,19,20,21,22,23,24,25,26,27,28,29,30,31);
    }
    #pragma unroll
    for (int i = 0; i < 4; ++i)
        #pragma unroll
        for (int j = 0; j < 2; ++j)
            acc[i][j] = __builtin_amdgcn_swmmac_f32_16x16x64_f16(
                false, afrag[i], false, bfrag[j],
                acc[i][j], (int)idxreg[i], false, false);
}

__global__ __launch_bounds__(256)
void g24_swmmac_gemm(const _Float16* __restrict__ wcomp,
                     const unsigned* __restrict__ widx,
                     const _Float16* __restrict__ xq,
                     const float* __restrict__ bias,
                     float* __restrict__ out) {
    __shared__ _Float16 Ws[2][TN][40];     // compressed W panels
    __shared__ _Float16 Xs[2][TMX][72];    // x panels
    __shared__ unsigned Is[2][TN][2];      // index words

    const int tid   = threadIdx.x;
    const int lane  = tid & 31;
    const int wave  = tid >> 5;
    const int waveA = wave >> 2;        // 0..1  (64-feature slabs)
    const int waveB = wave & 3;         // 0..3  (32-xrow slabs)
    const int m16   = lane & 15;
    const int hi    = lane >> 4;

    const int tileM = blockIdx.x * TMX;
    const int tileN = blockIdx.y * TN;

    const int ldr = tid >> 1;           // staging row 0..127
    const int sw  = tid & 1;

    v8f acc[4][2] = {};

    // prologue: panel 0 -> buffer 0
    stage_async(wcomp, widx, xq, Ws[0], Is[0], Xs[0], tileN, tileM, 0, ldr, sw);

    for (int k0 = 0; k0 < IN_F; k0 += 2 * TK) {
        // ---- phase A: buffer 0 resident, fill buffer 1 ----
        stage_async(wcomp, widx, xq, Ws[1], Is[1], Xs[1],
                    tileN, tileM, k0 + TK, ldr, sw);   // k0+TK < IN_F always
        WAIT_ASYNC_7();                 // own panel-A copies retired (in-order)
        __syncthreads();
        compute_panel(Ws[0], Is[0], Xs[0], waveA, waveB, m16, hi, acc);
        __syncthreads();                // all waves done reading buffer 0

        // ---- phase B: buffer 1 resident, refill buffer 0 ----
        if (k0 + 2 * TK < IN_F) {
            stage_async(wcomp, widx, xq, Ws[0], Is[0], Xs[0],
                        tileN, tileM, k0 + 2 * TK, ldr, sw);
            WAIT_ASYNC_7();
        } else {
            WAIT_ASYNC_0();
        }
        __syncthreads();
        compute_panel(Ws[1], Is[1], Xs[1], waveA, waveB, m16, hi, acc);
        __syncthreads();                // all waves done reading buffer 1
    }

    // ---- epilogue: D row = feature n (r + 8*hi), D col = x-row (m16) ----
    #pragma unroll
    for (int i = 0; i < 4; ++i) {
        int nBase = tileN + waveA * 64 + i * 16 + hi * 8;
        float4 bl = *reinterpret_cast<const float4*>(bias + nBase);
        float4 bh = *reinterpret_cast<const float4*>(bias + nBase + 4);
        #pragma unroll
        for (int j = 0; j < 2; ++j) {
            int m = tileM + waveB * 32 + j * 16 + m16;
            float4 o0 = make_float4(acc[i][j][0] + bl.x, acc[i][j][1] + bl.y,
                                    acc[i][j][2] + bl.z, acc[i][j][3] + bl.w);
            float4 o1 = make_float4(acc[i][j][4] + bh.x, acc[i][j][5] + bh.y,
                                    acc[i][j][6] + bh.z, acc[i][j][7] + bh.w);
            float4* dst = reinterpret_cast<float4*>(out + (size_t)m * OUT_F + nBase);
            dst[0] = o0;
            dst[1] = o1;
        }
    }
}

// ---------------------------------------------------------------------------
extern "C" void kernel_launch(void* const* d_in, const int* in_sizes, int n_in,
                              void* d_out, int out_size, void* d_ws, size_t ws_size,
                              hipStream_t stream) {
    const float* x    = (const float*)d_in[0];   // [2,2048,4096]
    const float* w    = (const float*)d_in[1];   // [4096,4096]
    const float* bias = (const float*)d_in[2];   // [4096]
    const float* cw   = (const float*)d_in[3];   // [N_BLOCKS,6]
    const float* gum  = (const float*)d_in[4];   // [N_BLOCKS,6]
    float* out        = (float*)d_out;           // [2,2048,4096]

    _Float16* wcomp = (_Float16*)d_ws;                          // 16 MB
    unsigned* widx  = (unsigned*)(wcomp + (size_t)OUT_F * KC);  // 2 MB
    _Float16* xq    = (_Float16*)(widx + NIDX32);               // 32 MB

    g24_pack_w<<<NIDX32 / 256, 256, 0, stream>>>(w, cw, gum, wcomp, widx);
    g24_quant_x<<<(M_TOT * IN_F / 4) / 256, 256, 0, stream>>>(x, xq);

    dim3 grid(M_TOT / TMX, OUT_F / TN);
    g24_swmmac_gemm<<<grid, 256, 0, stream>>>(wcomp, widx, xq, bias, out);
}